// Conv_13778255086166
// MI455X (gfx1250) — compile-verified
//
#include <hip/hip_runtime.h>
#include <math.h>

typedef float v2f __attribute__((ext_vector_type(2)));
typedef float v8f __attribute__((ext_vector_type(8)));

#define HO 64
#define INCH 64

// ---------------------------------------------------------------- zero scratch
__global__ __launch_bounds__(256) void k_zero(float* __restrict__ p, long long n) {
  long long i = (long long)blockIdx.x * blockDim.x + threadIdx.x;
  if (i < n) p[i] = 0.0f;
}

// ---------------------------------------------------------------- Xp = X @ W (f32 WMMA)
// One wave computes a 16x64 output tile: 16 K-steps of V_WMMA_F32_16X16X4_F32
// over 4 column tiles => 64 wmma / wave.
__global__ __launch_bounds__(256) void k_gemm_wmma(
    const float* __restrict__ X, const float* __restrict__ W,
    float* __restrict__ Xp, int nrows) {
  const int wave = (int)(((unsigned)blockIdx.x * blockDim.x + threadIdx.x) >> 5);
  const int lane = (int)(threadIdx.x & 31u);
  const int row0 = wave * 16;
  if (row0 >= nrows) return;                       // wave-uniform exit
  if (row0 + 16 <= nrows) {                        // wave-uniform: EXEC all-1s inside
    const int m  = lane & 15;                      // row (A) / col (B,C)
    const int kh = lane >> 4;                      // K half-select per ISA layout
    v8f acc0 = {}, acc1 = {}, acc2 = {}, acc3 = {};
#pragma unroll
    for (int k0 = 0; k0 < INCH; k0 += 4) {
      // A 16x4: lanes 0-15 hold K=k0+0,1 ; lanes 16-31 hold K=k0+2,3
      v2f a = *(const v2f*)(X + (size_t)(row0 + m) * INCH + (k0 + kh * 2));
      const float* w0 = W + (size_t)(k0 + kh * 2) * HO;  // B row K=k0+2*kh
      const float* w1 = w0 + HO;                         // B row K=k0+2*kh+1
      v2f b0, b1, b2, b3;
      b0.x = w0[m];      b0.y = w1[m];
      b1.x = w0[m + 16]; b1.y = w1[m + 16];
      b2.x = w0[m + 32]; b2.y = w1[m + 32];
      b3.x = w0[m + 48]; b3.y = w1[m + 48];
      acc0 = __builtin_amdgcn_wmma_f32_16x16x4_f32(false, a, false, b0, (short)0, acc0, false, false);
      acc1 = __builtin_amdgcn_wmma_f32_16x16x4_f32(false, a, false, b1, (short)0, acc1, false, false);
      acc2 = __builtin_amdgcn_wmma_f32_16x16x4_f32(false, a, false, b2, (short)0, acc2, false, false);
      acc3 = __builtin_amdgcn_wmma_f32_16x16x4_f32(false, a, false, b3, (short)0, acc3, false, false);
    }
    // C/D layout: VGPR r -> row r (lanes 0-15) / row 8+r (lanes 16-31), col = lane&15
#pragma unroll
    for (int r = 0; r < 8; ++r) {
      const int row = row0 + r + kh * 8;
      float* o = Xp + (size_t)row * HO + m;
      o[0]  = acc0[r];
      o[16] = acc1[r];
      o[32] = acc2[r];
      o[48] = acc3[r];
    }
  } else {
    // scalar tail (never hit for N=100000, kept for robustness; wave-uniform path)
    for (int r = row0 + lane; r < nrows; r += 32)
      for (int c = 0; c < HO; ++c) {
        float s = 0.f;
        for (int k = 0; k < INCH; ++k) s += X[(size_t)r * INCH + k] * W[(size_t)k * HO + c];
        Xp[(size_t)r * HO + c] = s;
      }
  }
}

// ---------------------------------------------------------------- pass 1: node -> edge
// thread t handles (nz = t>>4, channels 4*(t&15)..+3); also cnt / att_sum once per nz
__global__ __launch_bounds__(256) void k_scatter_v2e(
    const float* __restrict__ Xp, const int* __restrict__ vertex,
    const int* __restrict__ edges, const float* __restrict__ homo,
    float* __restrict__ Xe, float* __restrict__ cnt,
    float* __restrict__ att_sum, int nnz) {
  long long t = (long long)blockIdx.x * blockDim.x + threadIdx.x;
  int i  = (int)(t >> 4);
  int c4 = ((int)t & 15) * 4;
  if (i >= nnz) return;
  const int v = vertex[i];
  const int e = edges[i];
  const float4 x = *(const float4*)(Xp + (size_t)v * HO + c4);
  float* dst = Xe + (size_t)e * HO + c4;
  atomicAdd(dst + 0, x.x);
  atomicAdd(dst + 1, x.y);
  atomicAdd(dst + 2, x.z);
  atomicAdd(dst + 3, x.w);
  if (c4 == 0) {
    atomicAdd(cnt + e, 1.0f);
    atomicAdd(att_sum + v, homo[e]);
  }
}

// ---------------------------------------------------------------- Xe /= max(cnt,1)
__global__ __launch_bounds__(256) void k_edge_mean(float* __restrict__ Xe,
                                                   const float* __restrict__ cnt,
                                                   long long total) {
  long long i = (long long)blockIdx.x * blockDim.x + threadIdx.x;
  if (i >= total) return;
  const float c = cnt[i >> 6];
  Xe[i] *= 1.0f / fmaxf(c, 1.0f);
}

// ---------------------------------------------------------------- pass 2: edge -> node
__global__ __launch_bounds__(256) void k_scatter_e2v(
    const float* __restrict__ Xe, const int* __restrict__ vertex,
    const int* __restrict__ edges, const float* __restrict__ homo,
    const float* __restrict__ att_sum, float* __restrict__ out, int nnz) {
  long long t = (long long)blockIdx.x * blockDim.x + threadIdx.x;
  int i  = (int)(t >> 4);
  int c4 = ((int)t & 15) * 4;
  if (i >= nnz) return;
  const int v = vertex[i];
  const int e = edges[i];
  const float att = homo[e] / att_sum[v];
  const float4 x = *(const float4*)(Xe + (size_t)e * HO + c4);
  float* dst = out + (size_t)v * HO + c4;
  atomicAdd(dst + 0, att * x.x);
  atomicAdd(dst + 1, att * x.y);
  atomicAdd(dst + 2, att * x.z);
  atomicAdd(dst + 3, att * x.w);
}

// ---------------------------------------------------------------- row L2 normalize (wave32 per row)
__global__ __launch_bounds__(256) void k_rownorm(float* __restrict__ out, int n) {
  const int wave = (int)(((unsigned)blockIdx.x * blockDim.x + threadIdx.x) >> 5);
  const int lane = (int)(threadIdx.x & 31u);
  if (wave >= n) return;
  float* row = out + (size_t)wave * HO + lane * 2;
  float2 x = *(float2*)row;
  float ss = x.x * x.x + x.y * x.y;
#pragma unroll
  for (int off = 16; off > 0; off >>= 1) ss += __shfl_xor(ss, off, 32);
  const float nrm = sqrtf(ss);
  const float scale = (nrm > 0.f) ? 1.0f / fmaxf(nrm, 1e-30f) : 0.0f;
  x.x *= scale;
  x.y *= scale;
  *(float2*)row = x;
}

// ----------------------------------------------------------------
extern "C" void kernel_launch(void* const* d_in, const int* in_sizes, int n_in,
                              void* d_out, int out_size, void* d_ws, size_t ws_size,
                              hipStream_t stream) {
  const float* X      = (const float*)d_in[0];
  const float* W      = (const float*)d_in[1];
  const float* homo   = (const float*)d_in[2];
  const int*   vertex = (const int*)d_in[3];
  const int*   edges  = (const int*)d_in[4];
  float* out = (float*)d_out;

  const int n   = in_sizes[0] / INCH;   // 100000
  const int e   = in_sizes[2];          // 25000
  const int nnz = in_sizes[3];          // 2000000

  // workspace: Xe[e*64] | cnt[e] | att_sum[n]   (~6.9 MB, all L2-resident)
  float* Xe      = (float*)d_ws;
  float* cnt     = Xe + (size_t)e * HO;
  float* att_sum = cnt + e;
  const long long zn = (long long)e * HO + e + n;
  k_zero<<<(int)((zn + 255) / 256), 256, 0, stream>>>(Xe, zn);

  // Xp -> d_out (d_out doubles as Xp; pass 2 accumulates on top of it)
  const int tiles = (n + 15) / 16;
  k_gemm_wmma<<<(tiles * 32 + 255) / 256, 256, 0, stream>>>(X, W, out, n);

  const long long st = (long long)nnz * 16;
  k_scatter_v2e<<<(int)((st + 255) / 256), 256, 0, stream>>>(
      out, vertex, edges, homo, Xe, cnt, att_sum, nnz);

  const long long em = (long long)e * HO;
  k_edge_mean<<<(int)((em + 255) / 256), 256, 0, stream>>>(Xe, cnt, em);

  k_scatter_e2v<<<(int)((st + 255) / 256), 256, 0, stream>>>(
      Xe, vertex, edges, homo, att_sum, out, nnz);

  const long long nt = (long long)n * 32;
  k_rownorm<<<(int)((nt + 255) / 256), 256, 0, stream>>>(out, n);
}